// PFNet7_38165079392812
// MI455X (gfx1250) — compile-verified
//
#include <hip/hip_runtime.h>
#include <hip/hip_bf16.h>

// ---------------------------------------------------------------------------
// PFNet7 forward on MI455X (gfx1250).
// Dense layers use V_WMMA_F32_16X16X4_F32 (full f32 precision, matrix pipes).
// kNN top-16 is an LDS-tiled register scan. All scratch lives in d_ws.
// ---------------------------------------------------------------------------

typedef __attribute__((ext_vector_type(2))) float v2f;
typedef __attribute__((ext_vector_type(8))) float v8f;

#define SLOPE 0.01f

// ---------------------------------------------------------------------------
// Generic f32 WMMA GEMM:  Y[row,col] = act( A[M,K] * W^T + bias + addend )
//   A: lda-strided, rows = gridDim.x*16, K <= lda, K % 4 == 0, lda even
//   W: [colsTotal, ldw] row-major (so B[k][n] = W[n][k]), ldw even
//   One wave computes one 16x16 output tile; wave w handles cols [16w,16w+16).
//   blockDim.x = 32 * (#col tiles). No divergence before WMMA (EXEC all ones).
// ---------------------------------------------------------------------------
__global__ void wmma_gemm_f32(const float* __restrict__ A, int lda,
                              const float* __restrict__ W, int ldw,
                              const float* __restrict__ bias,
                              const float* __restrict__ addend, int ldadd,
                              float* __restrict__ Y, int ldy,
                              int K, int act)
{
    const int wave = threadIdx.x >> 5;
    const int lane = threadIdx.x & 31;
    const int m    = lane & 15;
    const int half = lane >> 4;
    const int row0 = blockIdx.x << 4;
    const int n0   = wave << 4;

    // Fragment base pointers (ISA 16x4 f32 A layout / 4x16 f32 B layout):
    //   lane L, vgpr v -> k = kb + 2*(L>>4) + v   (consecutive -> float2 load)
    const float* Ap = A + (size_t)(row0 + m) * lda + 2 * half;
    const float* Wp = W + (size_t)(n0 + m) * ldw + 2 * half;

    v8f acc = {0.f, 0.f, 0.f, 0.f, 0.f, 0.f, 0.f, 0.f};
    for (int kb = 0; kb < K; kb += 4) {
        v2f a = *(const v2f*)(Ap + kb);
        v2f b = *(const v2f*)(Wp + kb);
        acc = __builtin_amdgcn_wmma_f32_16x16x4_f32(
            /*neg_a=*/false, a, /*neg_b=*/false, b,
            /*c_mod=*/(short)0, acc, /*reuse_a=*/false, /*reuse_b=*/false);
    }

    // C/D layout: lane L -> col = L&15 ; vgpr r -> row = r + 8*(L>>4)
    const int col = n0 + m;
    const float bv = bias ? bias[col] : 0.0f;
#pragma unroll
    for (int r = 0; r < 8; ++r) {
        const int row = row0 + r + 8 * half;
        float v = acc[r] + bv;
        if (addend) v += addend[(size_t)row * ldadd + col];
        if (act) v = (v >= 0.f) ? v : SLOPE * v;
        Y[(size_t)row * ldy + col] = v;
    }
}

// ---------------------------------------------------------------------------
// Workspace prep: zero-fill + 2D pack (zero-padded, even-stride weights)
// ---------------------------------------------------------------------------
__global__ void zero_f32(float* __restrict__ p, int n)
{
    int i = blockIdx.x * blockDim.x + threadIdx.x;
    if (i < n) p[i] = 0.0f;
}

__global__ void pack2d(const float* __restrict__ src, float* __restrict__ dst,
                       int rows, int cols, int ldDst)
{
    int i = blockIdx.x * blockDim.x + threadIdx.x;
    if (i >= rows * cols) return;
    int r = i / cols, c = i - r * cols;
    dst[(size_t)r * ldDst + c] = src[i];
}

// ---------------------------------------------------------------------------
// s / ||s||^2 extraction from the fused [s|hf] activation buffer SH[N,32]
// ---------------------------------------------------------------------------
__global__ void prep_s(const float* __restrict__ SH, float4* __restrict__ s4,
                       float* __restrict__ sq, int N)
{
    int i = blockIdx.x * blockDim.x + threadIdx.x;
    if (i >= N) return;
    float a = SH[(size_t)i * 32 + 0], b = SH[(size_t)i * 32 + 1];
    float c = SH[(size_t)i * 32 + 2], d = SH[(size_t)i * 32 + 3];
    float4 v; v.x = a; v.y = b; v.z = c; v.w = d;
    s4[i] = v;
    sq[i] = a * a + b * b + c * c + d * d;
}

// ---------------------------------------------------------------------------
// kNN (K=16) over 4-dim learned coordinates; d2 = |si|^2 + |sj|^2 - 2 si.sj
// One query per thread, LDS tiles of 256 candidates, register top-16.
// ---------------------------------------------------------------------------
#define KNN_TILE 256
__global__ void knn16(const float4* __restrict__ s4, const float* __restrict__ sq,
                      int* __restrict__ idxO, float* __restrict__ wO, int N)
{
    __shared__ float4 ss[KNN_TILE];
    __shared__ float  ssq[KNN_TILE];

    const int i = blockIdx.x * blockDim.x + threadIdx.x;
    const float4 si = s4[i];
    const float sqi = sq[i];

    float bd[16]; int bi[16];
#pragma unroll
    for (int t = 0; t < 16; ++t) { bd[t] = 3.0e38f; bi[t] = i; }

    for (int jb = 0; jb < N; jb += KNN_TILE) {
        __syncthreads();
        ss[threadIdx.x]  = s4[jb + threadIdx.x];
        ssq[threadIdx.x] = sq[jb + threadIdx.x];
        __syncthreads();
#pragma unroll 4
        for (int jj = 0; jj < KNN_TILE; ++jj) {
            float4 sj = ss[jj];
            float dot = si.x * sj.x + si.y * sj.y + si.z * sj.z + si.w * sj.w;
            float d2  = sqi + ssq[jj] - 2.0f * dot;
            if (d2 < bd[15]) {               // replace current max, bubble down
                bd[15] = d2; bi[15] = jb + jj;
#pragma unroll
                for (int t = 15; t > 0; --t) {
                    if (bd[t] < bd[t - 1]) {
                        float td = bd[t]; bd[t] = bd[t - 1]; bd[t - 1] = td;
                        int   ti = bi[t]; bi[t] = bi[t - 1]; bi[t - 1] = ti;
                    }
                }
            }
        }
    }
#pragma unroll
    for (int t = 0; t < 16; ++t) {
        idxO[(size_t)i * 16 + t] = bi[t];
        wO[(size_t)i * 16 + t]   = __expf(-10.0f * bd[t]);
    }
}

// ---------------------------------------------------------------------------
// GravNet aggregation: mean & max over 16 neighbors of w*hf (22 feats),
// concat with h0 (12) into cat[N,64] (cols 56..63 zero pad).
// ---------------------------------------------------------------------------
__global__ void gravnet_agg(const float* __restrict__ SH, const float* __restrict__ h0p,
                            const int* __restrict__ idx, const float* __restrict__ w,
                            float* __restrict__ cat, int N)
{
    int i = blockIdx.x * blockDim.x + threadIdx.x;
    if (i >= N) return;
    float sum[22], mx[22];
#pragma unroll
    for (int c = 0; c < 22; ++c) { sum[c] = 0.f; mx[c] = -3.0e38f; }
    for (int k = 0; k < 16; ++k) {
        int   j  = idx[(size_t)i * 16 + k];
        float wk = w[(size_t)i * 16 + k];
        const float* hj = SH + (size_t)j * 32 + 4;
#pragma unroll
        for (int c = 0; c < 22; ++c) {
            float v = hj[c] * wk;
            sum[c] += v;
            mx[c] = fmaxf(mx[c], v);
        }
    }
    float* ci = cat + (size_t)i * 64;
#pragma unroll
    for (int c = 0; c < 22; ++c) { ci[c] = sum[c] * (1.0f / 16.0f); ci[22 + c] = mx[c]; }
#pragma unroll
    for (int c = 0; c < 12; ++c) ci[44 + c] = h0p[(size_t)i * 16 + c];
#pragma unroll
    for (int c = 56; c < 64; ++c) ci[c] = 0.f;
}

// ---------------------------------------------------------------------------
// GraphConv neighbor sum: nbr[i] = sum_k w[i,k] * x1[idx[i,k]]   (32 feats)
// ---------------------------------------------------------------------------
__global__ void graphconv_gather(const float* __restrict__ x1, const int* __restrict__ idx,
                                 const float* __restrict__ w, float* __restrict__ nbr, int N)
{
    int i = blockIdx.x * blockDim.x + threadIdx.x;
    if (i >= N) return;
    float acc[32];
#pragma unroll
    for (int c = 0; c < 32; ++c) acc[c] = 0.f;
    for (int k = 0; k < 16; ++k) {
        int   j  = idx[(size_t)i * 16 + k];
        float wk = w[(size_t)i * 16 + k];
        const float* xj = x1 + (size_t)j * 32;
#pragma unroll
        for (int c = 0; c < 32; ++c) acc[c] += xj[c] * wk;
    }
    float* o = nbr + (size_t)i * 32;
#pragma unroll
    for (int c = 0; c < 32; ++c) o[c] = acc[c];
}

// ---------------------------------------------------------------------------
// Finalize: write cand_ids to d_out, build cat2[N,40] = [x2(32) | ids(6) | 0 0]
// ---------------------------------------------------------------------------
__global__ void finalize_ids(const float* __restrict__ ids_p, const float* __restrict__ x2,
                             float* __restrict__ out_ids, float* __restrict__ cat2, int N)
{
    int i = blockIdx.x * blockDim.x + threadIdx.x;
    if (i >= N) return;
    float* c2 = cat2 + (size_t)i * 40;
#pragma unroll
    for (int c = 0; c < 32; ++c) c2[c] = x2[(size_t)i * 32 + c];
#pragma unroll
    for (int c = 0; c < 6; ++c) {
        float v = ids_p[(size_t)i * 16 + c];
        out_ids[(size_t)i * 6 + c] = v;
        c2[32 + c] = v;
    }
    c2[38] = 0.f; c2[39] = 0.f;
}

__global__ void finalize_p4(const float* __restrict__ p4_p, float* __restrict__ out_p4, int N)
{
    int i = blockIdx.x * blockDim.x + threadIdx.x;
    if (i >= N) return;
#pragma unroll
    for (int c = 0; c < 6; ++c) out_p4[(size_t)i * 6 + c] = p4_p[(size_t)i * 16 + c];
}

// ---------------------------------------------------------------------------
// Workspace layout (float offsets, 64-float aligned)
// ---------------------------------------------------------------------------
#define NMAX 16384
// packed weights
#define PW_NN1W0 0                        // [128,12]
#define PW_NN1W1 (PW_NN1W0 + 128*12)      // [128,128]
#define PW_NN1W2 (PW_NN1W1 + 128*128)     // [16,128]
#define PW_WSH   (PW_NN1W2 + 16*128)      // [32,12]  rows 0-3 Ws, 4-25 Wh
#define PW_NN2W0 (PW_WSH   + 32*12)       // [128,32]
#define PW_NN2W1 (PW_NN2W0 + 128*32)      // [128,128]
#define PW_NN2W2 (PW_NN2W1 + 128*128)     // [16,128]
#define PW_NN3W0 (PW_NN2W2 + 16*128)      // [128,40]
#define PW_NN3W1 (PW_NN3W0 + 128*40)      // [128,128]
#define PW_NN3W2 (PW_NN3W1 + 128*128)     // [16,128]
#define PB_NN1B0 (PW_NN3W2 + 16*128)      // [128]
#define PB_NN1B1 (PB_NN1B0 + 128)
#define PB_NN1B2 (PB_NN1B1 + 128)         // [64 used:16]
#define PB_SH    (PB_NN1B2 + 64)          // [64 used:32]
#define PB_NN2B0 (PB_SH    + 64)
#define PB_NN2B1 (PB_NN2B0 + 128)
#define PB_NN2B2 (PB_NN2B1 + 128)
#define PB_NN3B0 (PB_NN2B2 + 64)
#define PB_NN3B1 (PB_NN3B0 + 128)
#define PB_NN3B2 (PB_NN3B1 + 128)
#define PW_TOTAL (PB_NN3B2 + 64)
// activations
#define A_H1   ((PW_TOTAL + 63) & ~63)
#define A_H2   (A_H1  + NMAX*128)
#define A_H0P  (A_H2  + NMAX*128)         // [N,16]
#define A_SH   (A_H0P + NMAX*16)          // [N,32]  s(0..3) hf(4..25)
#define A_S4   (A_SH  + NMAX*32)          // [N,4]
#define A_SQ   (A_S4  + NMAX*4)           // [N]
#define A_IDX  (A_SQ  + NMAX)             // [N,16] int
#define A_WGT  (A_IDX + NMAX*16)          // [N,16]
#define A_CAT  (A_WGT + NMAX*16)          // [N,64]
#define A_X1   (A_CAT + NMAX*64)          // [N,32]
#define A_NBR  (A_X1  + NMAX*32)          // [N,32]
#define A_TMP  (A_NBR + NMAX*32)          // [N,32]
#define A_X2   (A_TMP + NMAX*32)          // [N,32]
#define A_IDS  (A_X2  + NMAX*32)          // [N,16]
#define A_CAT2 (A_IDS + NMAX*16)          // [N,40]
#define A_P4   (A_CAT2+ NMAX*40)          // [N,16]

extern "C" void kernel_launch(void* const* d_in, const int* in_sizes, int n_in,
                              void* d_out, int out_size, void* d_ws, size_t ws_size,
                              hipStream_t stream)
{
    // Input order per setup_inputs(): x, then params in dict insertion order.
    const float* x       = (const float*)d_in[0];
    const float* nn1_W0  = (const float*)d_in[1];
    const float* nn1_b0  = (const float*)d_in[2];
    const float* nn1_W1  = (const float*)d_in[3];
    const float* nn1_b1  = (const float*)d_in[4];
    const float* nn1_W2  = (const float*)d_in[5];
    const float* nn1_b2  = (const float*)d_in[6];
    const float* Ws      = (const float*)d_in[7];
    const float* bs      = (const float*)d_in[8];
    const float* Wh      = (const float*)d_in[9];
    const float* bh      = (const float*)d_in[10];
    const float* Wlin    = (const float*)d_in[11];
    const float* blin    = (const float*)d_in[12];
    const float* Wrel    = (const float*)d_in[13];
    const float* brel    = (const float*)d_in[14];
    const float* Wroot   = (const float*)d_in[15];
    const float* nn2_W0  = (const float*)d_in[16];
    const float* nn2_b0  = (const float*)d_in[17];
    const float* nn2_W1  = (const float*)d_in[18];
    const float* nn2_b1  = (const float*)d_in[19];
    const float* nn2_W2  = (const float*)d_in[20];
    const float* nn2_b2  = (const float*)d_in[21];
    const float* nn3_W0  = (const float*)d_in[22];
    const float* nn3_b0  = (const float*)d_in[23];
    const float* nn3_W1  = (const float*)d_in[24];
    const float* nn3_b1  = (const float*)d_in[25];
    const float* nn3_W2  = (const float*)d_in[26];
    const float* nn3_b2  = (const float*)d_in[27];

    const int N = in_sizes[0] / 12;          // 16384 (multiple of 256)
    float* ws = (float*)d_ws;
    float* out = (float*)d_out;

    auto gpk = [](int n) { return (n + 255) / 256; };

    // -------- 1. zero packed-weight region, pack padded weights/biases ------
    zero_f32<<<gpk(PW_TOTAL), 256, 0, stream>>>(ws, PW_TOTAL);
    pack2d<<<gpk(125*12), 256, 0, stream>>>(nn1_W0, ws + PW_NN1W0, 125, 12, 12);
    pack2d<<<gpk(125*125),256, 0, stream>>>(nn1_W1, ws + PW_NN1W1, 125, 125, 128);
    pack2d<<<gpk(12*125), 256, 0, stream>>>(nn1_W2, ws + PW_NN1W2, 12, 125, 128);
    pack2d<<<gpk(4*12),   256, 0, stream>>>(Ws,     ws + PW_WSH,          4, 12, 12);
    pack2d<<<gpk(22*12),  256, 0, stream>>>(Wh,     ws + PW_WSH + 4*12,  22, 12, 12);
    pack2d<<<gpk(125*32), 256, 0, stream>>>(nn2_W0, ws + PW_NN2W0, 125, 32, 32);
    pack2d<<<gpk(125*125),256, 0, stream>>>(nn2_W1, ws + PW_NN2W1, 125, 125, 128);
    pack2d<<<gpk(6*125),  256, 0, stream>>>(nn2_W2, ws + PW_NN2W2, 6, 125, 128);
    pack2d<<<gpk(125*38), 256, 0, stream>>>(nn3_W0, ws + PW_NN3W0, 125, 38, 40);
    pack2d<<<gpk(125*125),256, 0, stream>>>(nn3_W1, ws + PW_NN3W1, 125, 125, 128);
    pack2d<<<gpk(6*125),  256, 0, stream>>>(nn3_W2, ws + PW_NN3W2, 6, 125, 128);
    pack2d<<<1, 256, 0, stream>>>(nn1_b0, ws + PB_NN1B0, 1, 125, 125);
    pack2d<<<1, 256, 0, stream>>>(nn1_b1, ws + PB_NN1B1, 1, 125, 125);
    pack2d<<<1, 256, 0, stream>>>(nn1_b2, ws + PB_NN1B2, 1, 12, 12);
    pack2d<<<1, 256, 0, stream>>>(bs,     ws + PB_SH,     1, 4, 4);
    pack2d<<<1, 256, 0, stream>>>(bh,     ws + PB_SH + 4, 1, 22, 22);
    pack2d<<<1, 256, 0, stream>>>(nn2_b0, ws + PB_NN2B0, 1, 125, 125);
    pack2d<<<1, 256, 0, stream>>>(nn2_b1, ws + PB_NN2B1, 1, 125, 125);
    pack2d<<<1, 256, 0, stream>>>(nn2_b2, ws + PB_NN2B2, 1, 6, 6);
    pack2d<<<1, 256, 0, stream>>>(nn3_b0, ws + PB_NN3B0, 1, 125, 125);
    pack2d<<<1, 256, 0, stream>>>(nn3_b1, ws + PB_NN3B1, 1, 125, 125);
    pack2d<<<1, 256, 0, stream>>>(nn3_b2, ws + PB_NN3B2, 1, 6, 6);

    const int gRows = N / 16;                 // 16-row tiles

    // -------- 2. nn1 encoder: 12 -> 125 -> 125 -> 12 (padded 128/16) --------
    wmma_gemm_f32<<<gRows, 256, 0, stream>>>(x, 12, ws + PW_NN1W0, 12,
        ws + PB_NN1B0, nullptr, 0, ws + A_H1, 128, 12, 1);
    wmma_gemm_f32<<<gRows, 256, 0, stream>>>(ws + A_H1, 128, ws + PW_NN1W1, 128,
        ws + PB_NN1B1, nullptr, 0, ws + A_H2, 128, 128, 1);
    wmma_gemm_f32<<<gRows, 32, 0, stream>>>(ws + A_H2, 128, ws + PW_NN1W2, 128,
        ws + PB_NN1B2, nullptr, 0, ws + A_H0P, 16, 128, 0);

    // -------- 3. fused [s | hf] projection: 12 -> 32 (s:4, hf:22) -----------
    wmma_gemm_f32<<<gRows, 64, 0, stream>>>(ws + A_H0P, 16, ws + PW_WSH, 12,
        ws + PB_SH, nullptr, 0, ws + A_SH, 32, 12, 0);
    prep_s<<<gpk(N), 256, 0, stream>>>(ws + A_SH, (float4*)(ws + A_S4), ws + A_SQ, N);

    // -------- 4. kNN top-16 + edge weights ----------------------------------
    knn16<<<N / KNN_TILE, KNN_TILE, 0, stream>>>((const float4*)(ws + A_S4),
        ws + A_SQ, (int*)(ws + A_IDX), ws + A_WGT, N);

    // -------- 5. GravNet aggregate + Wlin -----------------------------------
    gravnet_agg<<<gpk(N), 256, 0, stream>>>(ws + A_SH, ws + A_H0P,
        (const int*)(ws + A_IDX), ws + A_WGT, ws + A_CAT, N);
    wmma_gemm_f32<<<gRows, 64, 0, stream>>>(ws + A_CAT, 64, Wlin, 56,
        blin, nullptr, 0, ws + A_X1, 32, 56, 1);

    // -------- 6. GraphConv: Wrel @ nbr_sum + brel + Wroot @ x1 --------------
    graphconv_gather<<<gpk(N), 256, 0, stream>>>(ws + A_X1,
        (const int*)(ws + A_IDX), ws + A_WGT, ws + A_NBR, N);
    wmma_gemm_f32<<<gRows, 64, 0, stream>>>(ws + A_NBR, 32, Wrel, 32,
        brel, nullptr, 0, ws + A_TMP, 32, 32, 0);
    wmma_gemm_f32<<<gRows, 64, 0, stream>>>(ws + A_X1, 32, Wroot, 32,
        nullptr, ws + A_TMP, 32, ws + A_X2, 32, 32, 1);

    // -------- 7. nn2 head: 32 -> 125 -> 125 -> 6 ----------------------------
    wmma_gemm_f32<<<gRows, 256, 0, stream>>>(ws + A_X2, 32, ws + PW_NN2W0, 32,
        ws + PB_NN2B0, nullptr, 0, ws + A_H1, 128, 32, 1);
    wmma_gemm_f32<<<gRows, 256, 0, stream>>>(ws + A_H1, 128, ws + PW_NN2W1, 128,
        ws + PB_NN2B1, nullptr, 0, ws + A_H2, 128, 128, 1);
    wmma_gemm_f32<<<gRows, 32, 0, stream>>>(ws + A_H2, 128, ws + PW_NN2W2, 128,
        ws + PB_NN2B2, nullptr, 0, ws + A_IDS, 16, 128, 0);
    finalize_ids<<<gpk(N), 256, 0, stream>>>(ws + A_IDS, ws + A_X2,
        out, ws + A_CAT2, N);

    // -------- 8. nn3 head: 38 -> 125 -> 125 -> 6 ----------------------------
    wmma_gemm_f32<<<gRows, 256, 0, stream>>>(ws + A_CAT2, 40, ws + PW_NN3W0, 40,
        ws + PB_NN3B0, nullptr, 0, ws + A_H1, 128, 40, 1);
    wmma_gemm_f32<<<gRows, 256, 0, stream>>>(ws + A_H1, 128, ws + PW_NN3W1, 128,
        ws + PB_NN3B1, nullptr, 0, ws + A_H2, 128, 128, 1);
    wmma_gemm_f32<<<gRows, 32, 0, stream>>>(ws + A_H2, 128, ws + PW_NN3W2, 128,
        ws + PB_NN3B2, nullptr, 0, ws + A_P4, 16, 128, 0);
    finalize_p4<<<gpk(N), 256, 0, stream>>>(ws + A_P4, out + (size_t)N * 6, N);

    (void)n_in; (void)out_size; (void)ws_size;
}